// GCN_40200893890739
// MI455X (gfx1250) — compile-verified
//
#include <hip/hip_runtime.h>
#include <hip/hip_bf16.h>
#include <cstdint>

typedef __attribute__((ext_vector_type(2))) float v2f;
typedef __attribute__((ext_vector_type(8))) float v8f;

#define BN_EPS 1e-5f
#define LDP 132   // LDS row pitch in floats (128+4 pad -> conflict-free b64 frag reads)

// ---------------------------------------------------------------- utilities
__global__ void zero_f_kernel(float* __restrict__ p, long long n) {
    long long i = (long long)blockIdx.x * blockDim.x + threadIdx.x;
    if (i < n) p[i] = 0.0f;
}

__global__ void deg_accum_kernel(const long long* __restrict__ dst,
                                 float* __restrict__ deg, long long E) {
    long long i = (long long)blockIdx.x * blockDim.x + threadIdx.x;
    if (i < E) atomicAdd(&deg[dst[i]], 1.0f);
}

__global__ void invdeg_kernel(float* __restrict__ deg, int n) {
    int i = blockIdx.x * blockDim.x + threadIdx.x;
    if (i < n) deg[i] = 1.0f / fmaxf(deg[i], 1.0f);
}

// ------------------------------------------------- W -> B-fragment repack
// W is [128 x dout] row-major. Packed layout (float2 units):
//   Wp2[(tile*32 + k4)*32 + lane] = { W[4*k4+khalf][tile*16+ncol],
//                                     W[4*k4+khalf+1][tile*16+ncol] }
// with khalf=(lane>>4)*2, ncol=lane&15 — i.e. exactly the per-lane B fragment
// for V_WMMA_F32_16X16X4_F32 at k-step k4.  One coalesced b64 per lane.
__global__ void pack_w_kernel(const float* __restrict__ W, float* __restrict__ Wp,
                              int dout) {
    int idx = blockIdx.x * blockDim.x + threadIdx.x;
    int total = (dout >> 4) * 32 * 32;        // tiles * k4-steps * lanes
    if (idx >= total) return;
    int lane = idx & 31;
    int k4   = (idx >> 5) & 31;
    int tile = idx >> 10;
    int k    = 4 * k4 + (((lane >> 4) << 1));
    int col  = tile * 16 + (lane & 15);
    float2 v;
    v.x = W[(size_t)k * dout + col];
    v.y = W[(size_t)(k + 1) * dout + col];
    *reinterpret_cast<float2*>(Wp + (size_t)idx * 2) = v;
}

// --------------------------------------------------- edge scatter (E waves)
// one wave (32 lanes) per edge; each lane moves one float4 of the 128-f row
__global__ void scatter_kernel(const float* __restrict__ x,
                               const long long* __restrict__ src,
                               const long long* __restrict__ dst,
                               float* __restrict__ agg, long long E) {
    long long idx = (long long)blockIdx.x * blockDim.x + threadIdx.x;
    if (idx >= E * 32) return;
    long long e = idx >> 5;
    int g = (int)(idx & 31);
    long long s = src[e];      // broadcast load within the wave
    long long d = dst[e];
    const float4 v = *reinterpret_cast<const float4*>(x + s * 128 + g * 4);
    float* o = agg + d * 128 + g * 4;
    atomicAdd(o + 0, v.x);
    atomicAdd(o + 1, v.y);
    atomicAdd(o + 2, v.z);
    atomicAdd(o + 3, v.w);
}

// ------------------------------------------------- fused SAGE layer (WMMA)
// out[16 x DOUT] = (agg*invdeg) @ Wl + bl + xin @ Wr   (+ optional BN+ReLU)
// blockDim = 256 (8 waves); wave w owns col-tile w.  Wl/Wr are pre-packed.
template <int DOUT, bool BN_RELU>
__global__ void sage_gemm_kernel(const float* __restrict__ agg,
                                 const float* __restrict__ invdeg,
                                 const float* __restrict__ xin,
                                 const float* __restrict__ WlP,
                                 const float* __restrict__ bl,
                                 const float* __restrict__ WrP,
                                 const float* __restrict__ gamma,
                                 const float* __restrict__ beta,
                                 const float* __restrict__ mean,
                                 const float* __restrict__ var,
                                 float* __restrict__ out) {
    __shared__ float lA[16 * LDP];   // invdeg-normalized agg tile
    __shared__ float lX[16 * LDP];   // x tile
    const int row0 = blockIdx.x * 16;
    const int tid  = threadIdx.x;

    // cooperative stage: 2 x (16 x 128) floats, float4 granularity
    for (int i = tid; i < 512; i += 256) {
        int r  = i >> 5;               // tile row
        int c4 = (i & 31) << 2;        // column (multiple of 4)
        float inv = invdeg[row0 + r];
        float4 av = *reinterpret_cast<const float4*>(agg + (size_t)(row0 + r) * 128 + c4);
        av.x *= inv; av.y *= inv; av.z *= inv; av.w *= inv;
        *reinterpret_cast<float4*>(&lA[r * LDP + c4]) = av;
        *reinterpret_cast<float4*>(&lX[r * LDP + c4]) =
            *reinterpret_cast<const float4*>(xin + (size_t)(row0 + r) * 128 + c4);
    }
    __syncthreads();

    const int wave = tid >> 5;
    const int lane = tid & 31;
    if (wave >= (DOUT >> 4)) return;   // wave-uniform: EXEC stays all-1 for WMMA
    const int mrow  = lane & 15;           // A: M = lane%16
    const int khalf = (lane >> 4) << 1;    // upper half-wave holds K+2,K+3
    const int ncol  = lane & 15;           // B/C/D: N = lane%16

    // per-lane bases (computed once; loop offsets are compile-time immediates)
    const float* aBase  = &lA[mrow * LDP + khalf];
    const float* xBase  = &lX[mrow * LDP + khalf];
    const float2* blT = reinterpret_cast<const float2*>(WlP) + ((size_t)wave * 32) * 32 + lane;
    const float2* brT = reinterpret_cast<const float2*>(WrP) + ((size_t)wave * 32) * 32 + lane;

    v8f c = {};
#pragma unroll
    for (int k4 = 0; k4 < 32; ++k4) {      // (agg/deg) @ Wl
        v2f a = *reinterpret_cast<const v2f*>(aBase + 4 * k4);
        float2 bf = blT[k4 * 32];
        v2f b; b.x = bf.x; b.y = bf.y;
        c = __builtin_amdgcn_wmma_f32_16x16x4_f32(false, a, false, b,
                                                  (short)0, c, false, false);
    }
#pragma unroll
    for (int k4 = 0; k4 < 32; ++k4) {      // xin @ Wr (same accumulator)
        v2f a = *reinterpret_cast<const v2f*>(xBase + 4 * k4);
        float2 bf = brT[k4 * 32];
        v2f b; b.x = bf.x; b.y = bf.y;
        c = __builtin_amdgcn_wmma_f32_16x16x4_f32(false, a, false, b,
                                                  (short)0, c, false, false);
    }

    // epilogue per ISA C/D layout: VGPR r -> M = r (lanes 0-15) / r+8 (16-31)
    const int col   = (wave << 4) + ncol;
    const int rbase = (lane >> 4) << 3;
    float blv = bl[col];
    float gsc = 0.0f, bsh = 0.0f;
    if (BN_RELU) {
        gsc = gamma[col] * rsqrtf(var[col] + BN_EPS);
        bsh = beta[col] - mean[col] * gsc;
    }
#pragma unroll
    for (int r = 0; r < 8; ++r) {
        float v = c[r] + blv;
        if (BN_RELU) v = fmaxf(v * gsc + bsh, 0.0f);
        out[(size_t)(row0 + rbase + r) * DOUT + col] = v;
    }
}

// -------------------------------------------------- log_softmax over 64 cols
// one wave per row; lane handles cols {lane, lane+32}
__global__ void log_softmax64_kernel(float* __restrict__ out, int nrows) {
    int row  = blockIdx.x * (blockDim.x >> 5) + (threadIdx.x >> 5);
    int lane = threadIdx.x & 31;
    if (row >= nrows) return;
    float* p = out + (size_t)row * 64;
    float a = p[lane], b = p[lane + 32];
    float m = fmaxf(a, b);
    for (int off = 16; off; off >>= 1) m = fmaxf(m, __shfl_xor(m, off, 32));
    float s = expf(a - m) + expf(b - m);
    for (int off = 16; off; off >>= 1) s += __shfl_xor(s, off, 32);
    float lse = m + logf(s);
    p[lane]      = a - lse;
    p[lane + 32] = b - lse;
}

// ------------------------------------------------------------------- driver
extern "C" void kernel_launch(void* const* d_in, const int* in_sizes, int n_in,
                              void* d_out, int out_size, void* d_ws, size_t ws_size,
                              hipStream_t stream) {
    const int N = 50000;
    const long long E = 640000;

    const float*     x   = (const float*)d_in[0];
    const long long* ei  = (const long long*)d_in[1];
    const long long* src = ei;
    const long long* dst = ei + E;
    const float *Wl0 = (const float*)d_in[2],  *bl0 = (const float*)d_in[3],
                *Wr0 = (const float*)d_in[4];
    const float *Wl1 = (const float*)d_in[5],  *bl1 = (const float*)d_in[6],
                *Wr1 = (const float*)d_in[7];
    const float *Wl2 = (const float*)d_in[8],  *bl2 = (const float*)d_in[9],
                *Wr2 = (const float*)d_in[10];
    const float *g0 = (const float*)d_in[11], *b0 = (const float*)d_in[12],
                *m0 = (const float*)d_in[13], *v0 = (const float*)d_in[14];
    const float *g1 = (const float*)d_in[15], *b1 = (const float*)d_in[16],
                *m1 = (const float*)d_in[17], *v1 = (const float*)d_in[18];
    (void)in_sizes; (void)n_in; (void)out_size; (void)ws_size;

    // workspace (floats): invdeg[Npad] | agg[N*128] | h0[N*128] | h1[N*128]
    //                     | 4x packed 128x128 W | 2x packed 128x64 W
    float* ws     = (float*)d_ws;
    const long long Npad = 50048;
    float* invdeg = ws;
    float* agg    = ws + Npad;
    float* h0     = agg + (long long)N * 128;
    float* h1     = h0  + (long long)N * 128;
    float* Wl0P   = h1  + (long long)N * 128;
    float* Wr0P   = Wl0P + 128 * 128;
    float* Wl1P   = Wr0P + 128 * 128;
    float* Wr1P   = Wl1P + 128 * 128;
    float* Wl2P   = Wr1P + 128 * 128;
    float* Wr2P   = Wl2P + 128 * 64;
    float* logits = (float*)d_out;   // [N,64]

    const int B = 256;
    const long long aggN = (long long)N * 128;
    dim3 gN(((long long)N + B - 1) / B);
    dim3 gAgg((aggN + B - 1) / B);
    dim3 gE((E + B - 1) / B);
    dim3 gScat((E * 32 + B - 1) / B);
    dim3 gGemm(N / 16);              // 3125, exact
    dim3 gSm(N / 8);                 // 6250, exact
    dim3 gPack128(8 * 32 * 32 / B);  // 32 blocks
    dim3 gPack64(4 * 32 * 32 / B);   // 16 blocks

    // repack weights into WMMA B-fragment order (runs every call; tiny)
    pack_w_kernel<<<gPack128, B, 0, stream>>>(Wl0, Wl0P, 128);
    pack_w_kernel<<<gPack128, B, 0, stream>>>(Wr0, Wr0P, 128);
    pack_w_kernel<<<gPack128, B, 0, stream>>>(Wl1, Wl1P, 128);
    pack_w_kernel<<<gPack128, B, 0, stream>>>(Wr1, Wr1P, 128);
    pack_w_kernel<<<gPack64,  B, 0, stream>>>(Wl2, Wl2P, 64);
    pack_w_kernel<<<gPack64,  B, 0, stream>>>(Wr2, Wr2P, 64);

    // degree -> 1/max(deg,1)
    zero_f_kernel<<<gN, B, 0, stream>>>(invdeg, N);
    deg_accum_kernel<<<gE, B, 0, stream>>>(dst, invdeg, E);
    invdeg_kernel<<<gN, B, 0, stream>>>(invdeg, N);

    // ---- layer 0: x -> h0 (128->128, BN0+ReLU)
    zero_f_kernel<<<gAgg, B, 0, stream>>>(agg, aggN);
    scatter_kernel<<<gScat, B, 0, stream>>>(x, src, dst, agg, E);
    sage_gemm_kernel<128, true><<<gGemm, B, 0, stream>>>(agg, invdeg, x, Wl0P, bl0, Wr0P,
                                                         g0, b0, m0, v0, h0);
    // ---- layer 1: h0 -> h1 (128->128, BN1+ReLU)
    zero_f_kernel<<<gAgg, B, 0, stream>>>(agg, aggN);
    scatter_kernel<<<gScat, B, 0, stream>>>(h0, src, dst, agg, E);
    sage_gemm_kernel<128, true><<<gGemm, B, 0, stream>>>(agg, invdeg, h0, Wl1P, bl1, Wr1P,
                                                         g1, b1, m1, v1, h1);
    // ---- layer 2: h1 -> logits (128->64, no BN)
    zero_f_kernel<<<gAgg, B, 0, stream>>>(agg, aggN);
    scatter_kernel<<<gScat, B, 0, stream>>>(h1, src, dst, agg, E);
    sage_gemm_kernel<64, false><<<gGemm, B, 0, stream>>>(agg, invdeg, h1, Wl2P, bl2, Wr2P,
                                                         nullptr, nullptr, nullptr, nullptr,
                                                         logits);
    // ---- log_softmax in place on d_out
    log_softmax64_kernel<<<gSm, B, 0, stream>>>(logits, N);
}